// EmbeddingLayer_32839319945356
// MI455X (gfx1250) — compile-verified
//
#include <hip/hip_runtime.h>
#include <math.h>

// ---------------------------------------------------------------------------
// CDNA5 (gfx1250) molecular GNN+FFN pipeline, f32 WMMA (v_wmma_f32_16x16x4_f32)
// ---------------------------------------------------------------------------

typedef __attribute__((ext_vector_type(2))) float v2f;
typedef __attribute__((ext_vector_type(8))) float v8f;

#define NNODE_IN   64
#define NEDGE_IN   16
#define NNODE_OUT  96
#define NEDGE_OUT  64
#define MOL_D      160   // 96 + 64
#define FP_DIM     2048
#define FP_EMBED   128
#define X0_D       288   // 160 + 128
#define MPAD       512   // 500 rows padded to 32 tiles of 16
#define NROWS      500

// workspace layout (float offsets)
#define OFF_A    0                        // [512 x 64]
#define OFF_C    (OFF_A + MPAD*64)        // [512 x 16]
#define OFF_CNT  (OFF_C + MPAD*16)        // [512]
#define OFF_X0   (OFF_CNT + MPAD)         // [512 x 288]
#define OFF_ZROW (OFF_X0 + MPAD*X0_D)     // [2048] zero dummy row
#define OFF_H1   (OFF_ZROW + FP_DIM)      // [512 x 512]
#define OFF_H2   (OFF_H1 + MPAD*512)      // [512 x 512]
#define ZERO_CNT OFF_H1                   // zero A,C,cnt,X0,ZROW (H1/H2 overwritten)

__device__ __forceinline__ v8f wmma4(v2f a, v2f b, v8f c) {
  // D = A(16x4,f32) * B(4x16,f32) + C(16x16,f32)
  return __builtin_amdgcn_wmma_f32_16x16x4_f32(
      false, a, false, b, (short)0, c, false, false);
}

// ---------------------------------------------------------------------------
__global__ void zero_kernel(float* p, int n) {
  int i = blockIdx.x * blockDim.x + threadIdx.x;
  if (i < n) p[i] = 0.0f;
}

// ---------------------------------------------------------------------------
// One pass over edges: A[g] += node_feats[src], C[g] += edge_feats[e], cnt[g]++
// Accumulators are ~160KB -> L2 resident atomics.
__global__ void edge_accum_kernel(const float* __restrict__ node_feats,
                                  const float* __restrict__ edge_feats,
                                  const int* __restrict__ src,
                                  const int* __restrict__ dst,
                                  const int* __restrict__ gids,
                                  float* __restrict__ Aacc,
                                  float* __restrict__ Cacc,
                                  float* __restrict__ cnt, int E) {
  int stride = gridDim.x * blockDim.x;
  for (int e = blockIdx.x * blockDim.x + threadIdx.x; e < E; e += stride) {
    int s = src[e];
    int g = gids[dst[e]];
    atomicAdd(&cnt[g], 1.0f);
    const float4* ef = (const float4*)(edge_feats + (size_t)e * NEDGE_IN);
    float* Cg = Cacc + g * NEDGE_IN;
    #pragma unroll
    for (int j = 0; j < 4; ++j) {
      float4 v = ef[j];
      atomicAdd(&Cg[4*j+0], v.x); atomicAdd(&Cg[4*j+1], v.y);
      atomicAdd(&Cg[4*j+2], v.z); atomicAdd(&Cg[4*j+3], v.w);
    }
    const float4* nf = (const float4*)(node_feats + (size_t)s * NNODE_IN);
    float* Ag = Aacc + g * NNODE_IN;
    #pragma unroll
    for (int j = 0; j < 16; ++j) {
      float4 v = nf[j];
      atomicAdd(&Ag[4*j+0], v.x); atomicAdd(&Ag[4*j+1], v.y);
      atomicAdd(&Ag[4*j+2], v.z); atomicAdd(&Ag[4*j+3], v.w);
    }
  }
}

// ---------------------------------------------------------------------------
// fp = relu(BN(fp_vector @ W_fp + b_fp)) -> X0[:, 160:288].
// One wave per 16x32 output strip (2 accumulators); 32 row-tiles x 4 strips
// = 128 waves, K=2048.  Rows >= 500 read a zeroed dummy row (branchless).
__global__ void fp_gemm_bn_kernel(const float* __restrict__ fpv,
                                  const float* __restrict__ W,
                                  const float* __restrict__ bias,
                                  const float* __restrict__ gamma,
                                  const float* __restrict__ beta,
                                  const float* __restrict__ mean,
                                  const float* __restrict__ var,
                                  const float* __restrict__ zrow,
                                  float* __restrict__ X0) {
  int tid = threadIdx.x, lane = tid & 31;
  int wid = blockIdx.x * (blockDim.x >> 5) + (tid >> 5);  // 0..127
  int mt = wid >> 2, np = wid & 3;
  int row0 = mt << 4, col0 = np << 5;
  int m = lane & 15, h = lane >> 4, kh = h << 1, n = lane & 15;
  int rowm = row0 + m;
  const float* arow = (rowm < NROWS) ? (fpv + (size_t)rowm * FP_DIM) : zrow;
  v8f acc0 = {}, acc1 = {};
  for (int k = 0; k < FP_DIM; k += 4) {
    int kk = k + kh;
    v2f a = *(const v2f*)(arow + kk);
    const float* bp0 = W + (size_t)kk * FP_EMBED + col0 + n;
    const float* bp1 = bp0 + FP_EMBED;
    v2f b0, b1;
    b0.x = bp0[0];  b0.y = bp1[0];
    b1.x = bp0[16]; b1.y = bp1[16];
    acc0 = wmma4(a, b0, acc0);
    acc1 = wmma4(a, b1, acc1);
  }
  int c0 = col0 + n, c1 = c0 + 16;
  float s0 = rsqrtf(var[c0] + 1e-5f) * gamma[c0];
  float s1 = rsqrtf(var[c1] + 1e-5f) * gamma[c1];
  #pragma unroll
  for (int r = 0; r < 8; ++r) {
    int row = row0 + r + (h << 3);
    if (row < NROWS) {
      float v0 = (acc0[r] + bias[c0] - mean[c0]) * s0 + beta[c0];
      float v1 = (acc1[r] + bias[c1] - mean[c1]) * s1 + beta[c1];
      X0[(size_t)row * X0_D + MOL_D + c0] = fmaxf(v0, 0.0f);
      X0[(size_t)row * X0_D + MOL_D + c1] = fmaxf(v1, 0.0f);
    }
  }
}

// ---------------------------------------------------------------------------
// mol = softmax([A@W_mpnn + cnt*b_mpnn , C@W_edge + cnt*b_edge]) -> X0[:,0:160]
// Block: 10 waves. Waves 0-5: mpnn col-tiles (K=64). Waves 6-9: edge col-tiles
// (K=16). Tile results staged in LDS, then a per-row softmax over 160 cols.
__global__ void mol_softmax_kernel(const float* __restrict__ Aacc,
                                   const float* __restrict__ Cacc,
                                   const float* __restrict__ cnt,
                                   const float* __restrict__ Wm,
                                   const float* __restrict__ bm,
                                   const float* __restrict__ We,
                                   const float* __restrict__ be,
                                   float* __restrict__ X0) {
  __shared__ float molS[16 * MOL_D];
  int tid = threadIdx.x, lane = tid & 31;
  int wv = tid >> 5;                 // 0..9
  int row0 = blockIdx.x << 4;        // 0..496 (padded accumulators -> safe)
  int m = lane & 15, h = lane >> 4, kh = h << 1, n = lane & 15;
  v8f acc = {};
  if (wv < 6) {                      // node projection: [16x64] @ [64x96]
    int col0 = wv << 4;
    const float* arow = Aacc + (size_t)(row0 + m) * NNODE_IN;
    for (int k = 0; k < NNODE_IN; k += 4) {
      int kk = k + kh;
      v2f a = *(const v2f*)(arow + kk);
      v2f b;
      b.x = Wm[kk * NNODE_OUT + col0 + n];
      b.y = Wm[(kk + 1) * NNODE_OUT + col0 + n];
      acc = wmma4(a, b, acc);
    }
    float bb = bm[col0 + n];
    #pragma unroll
    for (int r = 0; r < 8; ++r) {
      int mr = r + (h << 3);
      molS[mr * MOL_D + col0 + n] = acc[r] + cnt[row0 + mr] * bb;
    }
  } else {                           // edge projection: [16x16] @ [16x64]
    int col0 = (wv - 6) << 4;
    const float* arow = Cacc + (size_t)(row0 + m) * NEDGE_IN;
    for (int k = 0; k < NEDGE_IN; k += 4) {
      int kk = k + kh;
      v2f a = *(const v2f*)(arow + kk);
      v2f b;
      b.x = We[kk * NEDGE_OUT + col0 + n];
      b.y = We[(kk + 1) * NEDGE_OUT + col0 + n];
      acc = wmma4(a, b, acc);
    }
    float bb = be[col0 + n];
    #pragma unroll
    for (int r = 0; r < 8; ++r) {
      int mr = r + (h << 3);
      molS[mr * MOL_D + NNODE_OUT + col0 + n] = acc[r] + cnt[row0 + mr] * bb;
    }
  }
  __syncthreads();
  if (tid < 16) {                    // per-row softmax over 160 features
    float* rowp = molS + tid * MOL_D;
    float mx = -INFINITY;
    for (int c = 0; c < MOL_D; ++c) mx = fmaxf(mx, rowp[c]);
    float s = 0.0f;
    for (int c = 0; c < MOL_D; ++c) { float e = __expf(rowp[c] - mx); rowp[c] = e; s += e; }
    float inv = 1.0f / s;
    int row = row0 + tid;
    if (row < NROWS)
      for (int c = 0; c < MOL_D; ++c)
        X0[(size_t)row * X0_D + c] = rowp[c] * inv;
  }
}

// ---------------------------------------------------------------------------
// Generic f32 WMMA GEMM: Out[512 x N] = act(A[512 x K] @ B[K x N] + bias).
// One wave per 32x32 macro-tile (2x2 accumulators): per k-step, 2 x b64 (A)
// + 4 x b32 (B) feed 4 WMMAs -> 1.5 VMEM per matrix op.
__global__ void gemm_bias_act_kernel(const float* __restrict__ A,
                                     const float* __restrict__ Bm,
                                     const float* __restrict__ bias,
                                     float* __restrict__ Out,
                                     int K, int N, int lda, int ldout,
                                     int store_rows, int do_relu) {
  int tid = threadIdx.x, lane = tid & 31;
  int wid = blockIdx.x * (blockDim.x >> 5) + (tid >> 5);
  int npt = N >> 5;                  // # of 32-col strips
  int mt = wid / npt, np = wid % npt;
  int row0 = mt << 5, col0 = np << 5;
  int m = lane & 15, h = lane >> 4, kh = h << 1, n = lane & 15;
  const float* arow0 = A + (size_t)(row0 + m) * lda;
  const float* arow1 = arow0 + (size_t)16 * lda;
  v8f acc00 = {}, acc01 = {}, acc10 = {}, acc11 = {};
  for (int k = 0; k < K; k += 4) {
    int kk = k + kh;
    v2f a0 = *(const v2f*)(arow0 + kk);
    v2f a1 = *(const v2f*)(arow1 + kk);
    const float* bp0 = Bm + (size_t)kk * N + col0 + n;
    const float* bp1 = bp0 + N;
    v2f b0, b1;
    b0.x = bp0[0];  b0.y = bp1[0];
    b1.x = bp0[16]; b1.y = bp1[16];
    acc00 = wmma4(a0, b0, acc00);
    acc01 = wmma4(a0, b1, acc01);
    acc10 = wmma4(a1, b0, acc10);
    acc11 = wmma4(a1, b1, acc11);
  }
  float bb0 = bias[col0 + n], bb1 = bias[col0 + 16 + n];
  #pragma unroll
  for (int r = 0; r < 8; ++r) {
    int mr = r + (h << 3);
    int rowA = row0 + mr;
    int rowB = row0 + 16 + mr;
    if (rowA < store_rows) {
      float v0 = acc00[r] + bb0;
      float v1 = acc01[r] + bb1;
      if (do_relu) { v0 = fmaxf(v0, 0.0f); v1 = fmaxf(v1, 0.0f); }
      Out[(size_t)rowA * ldout + col0 + n]      = v0;
      Out[(size_t)rowA * ldout + col0 + 16 + n] = v1;
    }
    if (rowB < store_rows) {
      float v0 = acc10[r] + bb0;
      float v1 = acc11[r] + bb1;
      if (do_relu) { v0 = fmaxf(v0, 0.0f); v1 = fmaxf(v1, 0.0f); }
      Out[(size_t)rowB * ldout + col0 + n]      = v0;
      Out[(size_t)rowB * ldout + col0 + 16 + n] = v1;
    }
  }
}

// ---------------------------------------------------------------------------
extern "C" void kernel_launch(void* const* d_in, const int* in_sizes, int n_in,
                              void* d_out, int out_size, void* d_ws, size_t ws_size,
                              hipStream_t stream) {
  const float* node_feats = (const float*)d_in[0];
  const float* edge_feats = (const float*)d_in[1];
  const float* fp_vector  = (const float*)d_in[2];
  const int*   src        = (const int*)d_in[3];
  const int*   dst        = (const int*)d_in[4];
  const int*   gids       = (const int*)d_in[5];
  const float* W_mpnn     = (const float*)d_in[6];
  const float* b_mpnn     = (const float*)d_in[7];
  const float* W_edge     = (const float*)d_in[8];
  const float* b_edge     = (const float*)d_in[9];
  const float* W_fp       = (const float*)d_in[10];
  const float* b_fp       = (const float*)d_in[11];
  const float* bn_gamma   = (const float*)d_in[12];
  const float* bn_beta    = (const float*)d_in[13];
  const float* bn_mean    = (const float*)d_in[14];
  const float* bn_var     = (const float*)d_in[15];
  const float* W_ffn0     = (const float*)d_in[16];
  const float* b_ffn0     = (const float*)d_in[17];
  const float* W_ffn1     = (const float*)d_in[18];
  const float* b_ffn1     = (const float*)d_in[19];
  const float* W_ffn2     = (const float*)d_in[20];
  const float* b_ffn2     = (const float*)d_in[21];

  const int E = in_sizes[3];

  float* ws   = (float*)d_ws;
  float* Aacc = ws + OFF_A;
  float* Cacc = ws + OFF_C;
  float* cnt  = ws + OFF_CNT;
  float* X0   = ws + OFF_X0;
  float* zrow = ws + OFF_ZROW;
  float* H1   = ws + OFF_H1;
  float* H2   = ws + OFF_H2;

  // 1) zero accumulators + X0 + dummy row (determinism under graph replay)
  zero_kernel<<<(ZERO_CNT + 255) / 256, 256, 0, stream>>>(ws, ZERO_CNT);

  // 2) single edge pass with L2-resident float atomics
  edge_accum_kernel<<<(E + 255) / 256, 256, 0, stream>>>(
      node_feats, edge_feats, src, dst, gids, Aacc, Cacc, cnt, E);

  // 3) fingerprint encoder: GEMM + BN + ReLU -> X0[:,160:288] (128 waves)
  fp_gemm_bn_kernel<<<32, 128, 0, stream>>>(
      fp_vector, W_fp, b_fp, bn_gamma, bn_beta, bn_mean, bn_var, zrow, X0);

  // 4) mol projections + softmax -> X0[:,0:160] (32 blocks x 10 waves)
  mol_softmax_kernel<<<32, 320, 0, stream>>>(
      Aacc, Cacc, cnt, W_mpnn, b_mpnn, W_edge, b_edge, X0);

  // 5) FFN0: [512,288]@[288,512] +b, ReLU  (16 x 16 macro-tiles = 256 waves)
  gemm_bias_act_kernel<<<32, 256, 0, stream>>>(
      X0, W_ffn0, b_ffn0, H1, X0_D, 512, X0_D, 512, MPAD, 1);

  // 6) FFN1: [512,512]@[512,512] +b, ReLU  (256 waves)
  gemm_bias_act_kernel<<<32, 256, 0, stream>>>(
      H1, W_ffn1, b_ffn1, H2, 512, 512, 512, 512, MPAD, 1);

  // 7) FFN2: [512,512]@[512,256] +b -> d_out[500,256]  (16x8 = 128 waves)
  gemm_bias_act_kernel<<<16, 256, 0, stream>>>(
      H2, W_ffn2, b_ffn2, (float*)d_out, 512, 256, 512, 256, NROWS, 0);
}